// MemoryLayer_52046413693350
// MI455X (gfx1250) — compile-verified
//
#include <hip/hip_runtime.h>

// MemoryLayer hash-gather for gfx1250 (wave32).
//   x:      (4,2048,1024) f32      tables: (64,65536,32) f32      out: (4,2048,2048) f32
// Memory-bound (~160MB @ 23.3 TB/s ≈ 7us). No matmul structure -> no WMMA; the
// CDNA5-specific paths used are global_prefetch_b8 (hide random 128B row gather
// latency behind the transcendental gate computation) and non-temporal stores
// (keep the 192MB L2 available for table rows instead of streamed output).

typedef __attribute__((ext_vector_type(4))) float f32x4;

constexpr int kChunk        = 16;   // TAU / elements per chunk
constexpr int kOutChunk     = 32;   // floats per table row (128 bytes)
constexpr int kInDim        = 1024;
constexpr int kOutDim       = 2048;
constexpr int kTokens       = 4 * 2048;
constexpr int kWavesPerBlk  = 8;    // 256 threads = 8 wave32

__global__ __launch_bounds__(256)
void memlayer_hash_gather(const float* __restrict__ x,
                          const float* __restrict__ tables,
                          float* __restrict__ out)
{
    const int lane  = threadIdx.x & 31;
    const int wave  = threadIdx.x >> 5;
    const int token = blockIdx.x * kWavesPerBlk + wave;

    const float* xt = x   + (size_t)token * kInDim;
    float*       ot = out + (size_t)token * kOutDim;

    const int subrow = lane >> 3;  // which of 4 rows this lane serves in gather
    const int sub    = lane & 7;   // float4 slot within a 32-float row

#pragma unroll
    for (int pass = 0; pass < 2; ++pass) {
        const int c = pass * 32 + lane;          // this lane's chunk id (0..63)

        // ---- load this lane's 16-float chunk (4 x float4, 64B contiguous) ----
        float v[kChunk];
        const f32x4* xv = (const f32x4*)(xt + c * kChunk);
#pragma unroll
        for (int j = 0; j < 4; ++j) {
            f32x4 q = xv[j];
            v[j*4+0] = q.x; v[j*4+1] = q.y; v[j*4+2] = q.z; v[j*4+3] = q.w;
        }

        // ---- 16-bit sign hash: element 0 -> bit 15 (matches 1<<(TAU-1-i)) ----
        unsigned h = 0u;
#pragma unroll
        for (int i = 0; i < kChunk; ++i)
            h = (h << 1) | (v[i] >= 0.0f ? 1u : 0u);

        // ---- prefetch the random 128B table row NOW (gfx1250 global_prefetch),
        //      then hide its latency behind the transcendental gate below ----
        const float* rowp = tables + (((size_t)c << 16) + (size_t)h) * kOutChunk;
        __builtin_prefetch(rowp,      0, 3);
        __builtin_prefetch(rowp + 16, 0, 3);   // second 64B of the row

        // ---- gate: p = exp( sum_i log_sigmoid(2*x_i) ) ----
        float s = 0.0f;
#pragma unroll
        for (int i = 0; i < kChunk; ++i)
            s += __logf(1.0f + __expf(-2.0f * v[i]));
        const float p = __expf(-s);

        // ---- cooperative gather: 8 lanes per row, 4 rows per iteration.
        //      Broadcast (h,p) via wave32 shuffles; reads & writes are fully
        //      coalesced 512B transactions. ----
#pragma unroll
        for (int it = 0; it < 8; ++it) {
            const int      cl = it * 4 + subrow;        // chunk-local 0..31
            const unsigned hc = __shfl(h, cl, 32);
            const float    pc = __shfl(p, cl, 32);
            const int      cg = pass * 32 + cl;         // global chunk 0..63

            f32x4 r = *((const f32x4*)(tables + (((size_t)cg << 16) + (size_t)hc) * kOutChunk) + sub);
            r *= pc;
            __builtin_nontemporal_store(r, (f32x4*)(ot + cg * kOutChunk) + sub);
        }
    }
}

extern "C" void kernel_launch(void* const* d_in, const int* in_sizes, int n_in,
                              void* d_out, int out_size, void* d_ws, size_t ws_size,
                              hipStream_t stream)
{
    (void)in_sizes; (void)n_in; (void)d_ws; (void)ws_size; (void)out_size;
    const float* x      = (const float*)d_in[0];
    const float* tables = (const float*)d_in[1];
    float*       out    = (float*)d_out;

    const int blocks = kTokens / kWavesPerBlk;   // 1024 blocks of 256 threads
    memlayer_hash_gather<<<blocks, 256, 0, stream>>>(x, tables, out);
}